// GIN_16252156248490
// MI455X (gfx1250) — compile-verified
//
#include <hip/hip_runtime.h>

#define DF 128
#define XS_STRIDE 132   // 16 rows * 132 floats: row%64 bank offset = 4*row -> conflict-free A reads

typedef __attribute__((ext_vector_type(2))) float v2f;
typedef __attribute__((ext_vector_type(4))) float v4f;
typedef __attribute__((ext_vector_type(8))) float v8f;

// Order-preserving float -> u32 key (so GLOBAL_ATOMIC_MAX_U32 == float max).
__device__ __forceinline__ unsigned key_of(float x) {
    unsigned u = __float_as_uint(x);
    return (u & 0x80000000u) ? ~u : (u | 0x80000000u);
}
// key 0 == "no edge arrived" sentinel -> 0.0f (reference: empty neighborhoods -> 0)
__device__ __forceinline__ float val_of(unsigned k) {
    if (k == 0u) return 0.0f;
    return (k & 0x80000000u) ? __uint_as_float(k ^ 0x80000000u)
                             : __uint_as_float(~k);
}

__global__ __launch_bounds__(256) void gin_fill_zero(unsigned* __restrict__ p, int n) {
    int i = blockIdx.x * 256 + threadIdx.x;
    if (i < n) p[i] = 0u;
}

// One wave32 per edge; each lane handles 4 consecutive features (float4 gather,
// 4 u32 atomic-max). Gather source (51.2 MB) lives in L2.
__global__ __launch_bounds__(256) void gin_scatter_max(const float* __restrict__ x,
                                                       const int* __restrict__ src,
                                                       const int* __restrict__ dst,
                                                       unsigned* __restrict__ aggk,
                                                       int nEdges) {
    int tid  = blockIdx.x * 256 + threadIdx.x;
    int e    = tid >> 5;
    int lane = tid & 31;
    if (e >= nEdges) return;
    int s = src[e];
    int d = dst[e];
    v4f v = *(const v4f*)(x + (size_t)s * DF + lane * 4);
    unsigned* a = aggk + (size_t)d * DF + lane * 4;
    atomicMax(a + 0, key_of(v.x));
    atomicMax(a + 1, key_of(v.y));
    atomicMax(a + 2, key_of(v.z));
    atomicMax(a + 3, key_of(v.w));
}

// out[i,:] = act((x[i,:] + agg[i,:]) @ W + b)
// Block = 256 threads = 8 waves. Block owns 16 node rows; wave w owns output
// columns [16w, 16w+16). K=128 consumed as 32x V_WMMA_F32_16X16X4_F32.
template <bool RELU>
__global__ __launch_bounds__(256) void gin_gemm_wmma(const float* __restrict__ x,
                                                     const unsigned* __restrict__ aggk,
                                                     const float* __restrict__ W,
                                                     const float* __restrict__ bias,
                                                     float* __restrict__ out,
                                                     int nNodes) {
    __shared__ float Xs[16][XS_STRIDE];

    const int t    = blockIdx.x;          // node tile: rows [16t, 16t+16)
    const int tid  = threadIdx.x;
    const int w    = tid >> 5;            // wave id 0..7 -> column tile
    const int lane = tid & 31;

    // Cooperatively build X = h + decode(aggmax) for this 16x128 tile.
    const size_t base     = (size_t)t * 16 * DF;
    const bool   tileFull = (t * 16 + 16) <= nNodes;   // uniform across block
    for (int i = tid; i < 16 * DF; i += 256) {
        int r = i >> 7, c = i & 127;
        float xv = 0.0f;
        if (tileFull || (t * 16 + r) < nNodes) {
            size_t idx = base + (size_t)r * DF + c;
            xv = x[idx] + val_of(aggk[idx]);
        }
        Xs[r][c] = xv;
    }
    __syncthreads();

    // Fragment coordinates per ISA layout:
    //  A 16x4 f32: lane L -> row = L&15, VGPR pair holds K = kbase..kbase+1,
    //              kbase = 2*(L>>4) within the 4-wide K slab.
    //  B 4x16 f32: lane L -> col = L&15, same K striping.
    //  C/D 16x16:  VGPR v -> row = v + 8*(L>>4), col = L&15.
    const int col  = lane & 15;
    const int koff = (lane >> 4) << 1;    // 0 or 2
    const int gcol = (w << 4) + col;

    // Bias depends on column only -> fold into accumulator init (saves epilogue adds).
    const float bc = bias[gcol];
    v8f acc = {bc, bc, bc, bc, bc, bc, bc, bc};

#pragma unroll 8
    for (int k0 = 0; k0 < DF; k0 += 4) {
        v2f a = *(const v2f*)&Xs[lane & 15][k0 + koff];       // conflict-free DS read
        v2f b;
        b.x = W[(size_t)(k0 + koff)     * DF + gcol];
        b.y = W[(size_t)(k0 + koff + 1) * DF + gcol];
        acc = __builtin_amdgcn_wmma_f32_16x16x4_f32(
            /*neg_a=*/false, a, /*neg_b=*/false, b,
            /*c_mod=*/(short)0, acc, /*reuse_a=*/false, /*reuse_b=*/false);
    }

    const int rbase = (lane >> 4) << 3;   // 0 or 8
    float* o = out + base;
    if (tileFull) {
        // Fast path (always taken for N % 16 == 0): unguarded stores, no exec churn.
#pragma unroll
        for (int v = 0; v < 8; ++v) {
            float r = acc[v];
            if (RELU) r = fmaxf(r, 0.0f);
            o[(size_t)(rbase + v) * DF + gcol] = r;
        }
    } else {
#pragma unroll
        for (int v = 0; v < 8; ++v) {
            if ((t * 16 + rbase + v) < nNodes) {
                float r = acc[v];
                if (RELU) r = fmaxf(r, 0.0f);
                o[(size_t)(rbase + v) * DF + gcol] = r;
            }
        }
    }
}

extern "C" void kernel_launch(void* const* d_in, const int* in_sizes, int n_in,
                              void* d_out, int out_size, void* d_ws, size_t ws_size,
                              hipStream_t stream) {
    const float* h   = (const float*)d_in[0];
    const int*   src = (const int*)d_in[1];
    const int*   dst = (const int*)d_in[2];
    const float* W1  = (const float*)d_in[3];
    const float* b1  = (const float*)d_in[4];
    const float* W2  = (const float*)d_in[5];
    const float* b2  = (const float*)d_in[6];
    float*       out = (float*)d_out;

    const int nNodes = in_sizes[0] / DF;
    const int nEdges = in_sizes[1];
    const int nFeat  = nNodes * DF;

    unsigned* aggk = (unsigned*)d_ws;                                  // N*128 u32 keys
    float*    xmid = (float*)((char*)d_ws + (size_t)nFeat * 4);        // layer-1 activations

    const int fillBlocks = (nFeat + 255) / 256;
    const int scatBlocks = (int)(((long long)nEdges * 32 + 255) / 256);
    const int tiles      = (nNodes + 15) / 16;

    // Layer 1: agg = segment_max(h[src], dst); x = relu((h+agg)@W1 + b1)
    gin_fill_zero  <<<fillBlocks, 256, 0, stream>>>(aggk, nFeat);
    gin_scatter_max<<<scatBlocks, 256, 0, stream>>>(h, src, dst, aggk, nEdges);
    gin_gemm_wmma<true><<<tiles, 256, 0, stream>>>(h, aggk, W1, b1, xmid, nNodes);

    // Layer 2: agg = segment_max(x[src], dst); out = (x+agg)@W2 + b2
    gin_fill_zero  <<<fillBlocks, 256, 0, stream>>>(aggk, nFeat);
    gin_scatter_max<<<scatBlocks, 256, 0, stream>>>(xmid, src, dst, aggk, nEdges);
    gin_gemm_wmma<false><<<tiles, 256, 0, stream>>>(xmid, aggk, W2, b2, out, nNodes);
}